// GCN_13632226197527
// MI455X (gfx1250) — compile-verified
//
#include <hip/hip_runtime.h>
#include <stdint.h>

// gspmmv: out[dst] += input[src] * weight   (scale fused into scatter, since
// (sum_e x_src) * w == sum_e (x_src * w))
//
// Shapes: input [N=50000, F=128] f32, weight [1,128] f32,
//         edge_src/edge_dst [E=600000] int32 (harness passes integers as int).
//
// Memory plan (MI455X): whole working set (~56 MB) lives in the 192 MB L2.
// Gather = 307 MB of L2 reads via async global->LDS b128 copies (double
// buffered, ASYNCcnt pipelined). Scatter = 76.8M f32 atomic adds executed at
// the L2 atomic units. Compute is negligible -> pure movement kernel.

#define D_FEAT 128
#define WAVES_PER_BLOCK 8
#define BLOCK_THREADS (WAVES_PER_BLOCK * 32)
#define SCATTER_BLOCKS 6000   // 48000 waves, ~12 edges each (grid-stride)

__global__ void __launch_bounds__(256) gcn_zero_kernel(float4* __restrict__ out, int n4) {
  int i = blockIdx.x * blockDim.x + threadIdx.x;
  if (i < n4) out[i] = make_float4(0.f, 0.f, 0.f, 0.f);
}

__global__ void __launch_bounds__(BLOCK_THREADS) gcn_scatter_kernel(
    const float* __restrict__ x,     // [N, 128]
    const float* __restrict__ w,     // [128]
    const int*   __restrict__ esrc,  // [E]
    const int*   __restrict__ edst,  // [E]
    float*       __restrict__ out,   // [N, 128]
    int n_edges) {
  // Per-wave double buffer: 2 x 512 B. 8 KiB per block (of 320 KiB/WGP).
  __shared__ float lds[WAVES_PER_BLOCK][2][D_FEAT];

  const int lane = threadIdx.x & 31;
  // Force the wave id scalar so edge-index loads become s_load (KMcnt).
  const int wid = __builtin_amdgcn_readfirstlane(threadIdx.x >> 5);

  // Each lane owns 4 consecutive features; 32 lanes cover all 128.
  const float4 wv = ((const float4*)w)[lane];

  const int nwaves = gridDim.x * WAVES_PER_BLOCK;
  int e = blockIdx.x * WAVES_PER_BLOCK + wid;   // wave-uniform
  if (e >= n_edges) return;

  // LDS byte offsets for this lane's 16B slot in each buffer
  // (flat-address low 32 bits == LDS offset on gfx1250).
  unsigned ldsoff0 = (unsigned)(uintptr_t)&lds[wid][0][lane * 4];
  unsigned ldsoff1 = (unsigned)(uintptr_t)&lds[wid][1][lane * 4];

  // Prime the pipeline: async gather of this wave's first source row.
  {
    int s = esrc[e];
    const float* g = x + (size_t)s * D_FEAT + lane * 4;
    asm volatile("global_load_async_to_lds_b128 %0, %1, off"
                 :: "v"(ldsoff0), "v"((uint64_t)(uintptr_t)g)
                 : "memory");
  }

  int p = 0;  // parity of the buffer holding edge e's row
  for (; e < n_edges; e += nwaves) {
    const int en = e + nwaves;        // wave-uniform condition
    if (en < n_edges) {
      // Issue next gather into the other buffer, then wait until only that
      // newly-issued op is outstanding (async loads complete in order).
      int s = esrc[en];
      const float* g = x + (size_t)s * D_FEAT + lane * 4;
      unsigned dst = p ? ldsoff0 : ldsoff1;
      asm volatile("global_load_async_to_lds_b128 %0, %1, off"
                   :: "v"(dst), "v"((uint64_t)(uintptr_t)g)
                   : "memory");
      asm volatile("s_wait_asynccnt 1" ::: "memory");
    } else {
      asm volatile("s_wait_asynccnt 0" ::: "memory");
    }

    // Consume current buffer: ds_load_b128, fuse weight scale, scatter with
    // f32 atomics (resolved at the L2 atomic units).
    const float4 v = *(const float4*)&lds[wid][p][lane * 4];
    const int d = edst[e];
    float* o = out + (size_t)d * D_FEAT + lane * 4;
    unsafeAtomicAdd(o + 0, v.x * wv.x);
    unsafeAtomicAdd(o + 1, v.y * wv.y);
    unsafeAtomicAdd(o + 2, v.z * wv.z);
    unsafeAtomicAdd(o + 3, v.w * wv.w);

    p ^= 1;
  }
}

extern "C" void kernel_launch(void* const* d_in, const int* in_sizes, int n_in,
                              void* d_out, int out_size, void* d_ws, size_t ws_size,
                              hipStream_t stream) {
  const float* x    = (const float*)d_in[0];
  const float* w    = (const float*)d_in[1];
  const int*   esrc = (const int*)d_in[2];
  const int*   edst = (const int*)d_in[3];
  float* out = (float*)d_out;

  const int n_edges = in_sizes[2];

  // 1) zero the accumulator (harness poisons d_out; we must re-zero per call)
  const int n4 = out_size / 4;
  gcn_zero_kernel<<<(n4 + 255) / 256, 256, 0, stream>>>((float4*)out, n4);

  // 2) fused gather + scale + atomic scatter
  int blocks = SCATTER_BLOCKS;
  const int need = (n_edges + WAVES_PER_BLOCK - 1) / WAVES_PER_BLOCK;
  if (blocks > need) blocks = need;
  gcn_scatter_kernel<<<blocks, BLOCK_THREADS, 0, stream>>>(x, w, esrc, edst, out, n_edges);
}